// FGN_Conv1d_84765474554312
// MI455X (gfx1250) — compile-verified
//
#include <hip/hip_runtime.h>

typedef float v2f __attribute__((ext_vector_type(2)));
typedef float v8f __attribute__((ext_vector_type(8)));

#define B_    8
#define CIN   32
#define COUT  64
#define KW    9
#define L_    512
#define PAD   4
#define T_    512
#define NT    64
#define SLICE (NT + 2 * PAD)   // 72
#define KDIM  (CIN * KW)       // 288
#define XPITCH (CIN + 2)       // 34 floats: even (8B-aligned rows) + conflict-free banks

// ---- prep: Wt[m][tap*32 + cin] = w[m][cin][tap]  (72 KB table in d_ws, L2-resident)
__global__ __launch_bounds__(256) void fgn_transpose_w(const float* __restrict__ w,
                                                       float* __restrict__ wt) {
    int i = blockIdx.x * 256 + threadIdx.x;
    if (i < COUT * KDIM) {
        int m = i / KDIM, r = i % KDIM;
        int cin = r / KW, tap = r % KW;
        wt[m * KDIM + tap * CIN + cin] = w[i];
    }
}

__global__ __launch_bounds__(256) void fgn_conv1d_wmma(
    const float* __restrict__ x,        // (B, CIN, L)
    const float* __restrict__ wt,       // (COUT, KDIM) transposed weights
    const float* __restrict__ bias,     // (COUT)
    const float* __restrict__ centers,  // (COUT, KW)
    const float* __restrict__ invc,     // (COUT, KW)
    float* __restrict__ out)            // [0..B*COUT*T) = c*g, [B*COUT*T..) = g
{
    __shared__ __align__(16) float s_xt[SLICE][XPITCH]; // transposed padded slice
    __shared__ float s_Q[SLICE];        // sum over cin of x^2
    __shared__ float s_P[SLICE];        // sum over cin of x
    __shared__ float s_A[COUT * KW];    // inv2
    __shared__ float s_Bc[COUT * KW];   // -2*inv2*center
    __shared__ float s_C0[COUT];        // 32 * sum_k inv2*center^2
    __shared__ float s_bias[COUT];

    const int tid = threadIdx.x;
    const int b   = blockIdx.x;
    const int t0  = blockIdx.y * NT;

    // ---- stage padded input slice (transposed): s_xt[j][c] = pad(x)[b, c, t0+j]
    const float* xb = x + (size_t)b * CIN * L_;
    for (int i = tid; i < CIN * SLICE; i += 256) {
        int c = i / SLICE, j = i % SLICE;          // contiguous global reads in j
        int src = t0 + j - PAD;
        s_xt[j][c] = (src >= 0 && src < L_) ? xb[c * L_ + src] : 0.0f;
    }

    // ---- gaussian coefficient tables
    for (int i = tid; i < COUT * KW; i += 256) {
        float sc = fminf(invc[i], 1e8f);
        float i2 = sc * sc;
        s_A[i]  = i2;
        s_Bc[i] = -2.0f * i2 * centers[i];
    }
    if (tid < COUT) s_bias[tid] = bias[tid];
    __syncthreads();

    if (tid < SLICE) {
        float p = 0.f, q = 0.f;
        #pragma unroll
        for (int c = 0; c < CIN; ++c) {
            float v = s_xt[tid][c];
            p += v; q += v * v;
        }
        s_P[tid] = p; s_Q[tid] = q;
    }
    if (tid < COUT) {
        float c0 = 0.f;
        #pragma unroll
        for (int k = 0; k < KW; ++k) {
            float ce = centers[tid * KW + k];
            c0 += s_A[tid * KW + k] * ce * ce;
        }
        s_C0[tid] = (float)CIN * c0;
    }
    __syncthreads();

    // ---- WMMA GEMM: C(64x64) = Wt(64x288) * X(288x64), kidx = tap*32 + cin
    const int wave = tid >> 5;            // 8 waves
    const int lane = tid & 31;
    const int half = lane >> 4;           // 0: K pair (kk,kk+1); 1: (kk+2,kk+3)
    const int l16  = lane & 15;
    const int m0   = (wave >> 1) * 16;    // 4 row-subtiles, 2 waves each (share A)
    const int n0   = (wave & 1) * 32;     // each wave owns columns n0..n0+31

    v8f acc0 = {};
    v8f acc1 = {};

    // this lane's weight row, viewed as 8-byte fragments (KDIM even, base aligned)
    const v2f* wrow2 = (const v2f*)(wt + (m0 + l16) * KDIM);

    for (int kk = 0; kk < KDIM; kk += 4) {
        const int tap  = kk >> 5;                 // 288 = 9 taps * 32 cin
        const int cinA = (kk & 31) + 2 * half;    // even -> 8B-aligned LDS b64

        // A fragment: Wt[m][kk+2*half], Wt[m][kk+2*half+1]  (one global_load_b64)
        v2f a = wrow2[(kk >> 1) + half];

        // B fragments: X[kidx, n] = s_xt[n + tap][cin]  (one ds_load_b64 each)
        v2f b0 = *(const v2f*)&s_xt[n0      + l16 + tap][cinA];
        v2f b1 = *(const v2f*)&s_xt[n0 + 16 + l16 + tap][cinA];

        acc0 = __builtin_amdgcn_wmma_f32_16x16x4_f32(false, a, false, b0,
                                                     (short)0, acc0, false, false);
        acc1 = __builtin_amdgcn_wmma_f32_16x16x4_f32(false, a, false, b1,
                                                     (short)0, acc1, false, false);
    }

    // ---- fused epilogue: bias, gaussian gate, exp, both outputs
    float* out1 = out;
    float* outg = out + (size_t)B_ * COUT * T_;
    v8f accs[2] = {acc0, acc1};

    #pragma unroll
    for (int s = 0; s < 2; ++s) {
        const int t_local = n0 + s * 16 + l16;    // C-tile: N = lane&15
        const int t = t0 + t_local;
        #pragma unroll
        for (int r = 0; r < 8; ++r) {
            const int m = m0 + r + 8 * half;      // C-tile: VGPR r -> M = r (+8 hi half)
            float cval = accs[s][r] + s_bias[m];
            float gs = s_C0[m];
            #pragma unroll
            for (int k = 0; k < KW; ++k) {
                gs += s_A[m * KW + k]  * s_Q[t_local + k]
                    + s_Bc[m * KW + k] * s_P[t_local + k];
            }
            float g = __expf(-gs);
            size_t o = ((size_t)b * COUT + m) * T_ + t;
            out1[o] = cval * g;
            outg[o] = g;
        }
    }
}

extern "C" void kernel_launch(void* const* d_in, const int* in_sizes, int n_in,
                              void* d_out, int out_size, void* d_ws, size_t ws_size,
                              hipStream_t stream) {
    const float* x       = (const float*)d_in[0];
    const float* w       = (const float*)d_in[1];
    const float* bias    = (const float*)d_in[2];
    const float* centers = (const float*)d_in[3];
    const float* invc    = (const float*)d_in[4];
    float*       wt      = (float*)d_ws;          // needs COUT*KDIM*4 = 73728 B

    // 1) transpose weights into workspace (same stream -> ordered before main)
    fgn_transpose_w<<<(COUT * KDIM + 255) / 256, 256, 0, stream>>>(w, wt);

    // 2) fused conv + gaussian-gate kernel: 8 batches x 8 t-tiles
    dim3 grid(B_, T_ / NT);
    fgn_conv1d_wmma<<<grid, 256, 0, stream>>>(x, wt, bias, centers, invc,
                                              (float*)d_out);
}